// GNN_4741643895562
// MI455X (gfx1250) — compile-verified
//
#include <hip/hip_runtime.h>

#define N_NODES 100000
#define N_EDGES 1200000
#define D 64

typedef __attribute__((ext_vector_type(2))) float v2f;
typedef __attribute__((ext_vector_type(8))) float v8f;

// ---------------------------------------------------------------------------
// Edge scatter: one wave32 per edge; lane l owns feature dims {2l, 2l+1}.
//   agg_f[dst] += x[src]   (forward aggregation)
//   agg_r[src] += x[dst]   (reverse aggregation)
// Edge index is wave-uniform: readfirstlane it so src/dst become scalar
// (SMEM) loads and row-base math lives in SGPRs.
// ---------------------------------------------------------------------------
__global__ __launch_bounds__(256) void sage_scatter(
    const float* __restrict__ xin,
    const int* __restrict__ src,
    const int* __restrict__ dst,
    float* __restrict__ agg_f,
    float* __restrict__ agg_r,
    float* __restrict__ cnt_f,
    float* __restrict__ cnt_r,
    int do_count)
{
    // 8 waves per block; one edge per wave (uniform across the wave)
    int e = __builtin_amdgcn_readfirstlane(
        (int)(blockIdx.x * 8 + (threadIdx.x >> 5)));
    if (e >= N_EDGES) return;
    int lane = threadIdx.x & 31;

    int s = src[e];   // uniform address -> scalar load
    int d = dst[e];

    const float2* xs = (const float2*)(xin + (size_t)s * D) + lane;
    const float2* xd = (const float2*)(xin + (size_t)d * D) + lane;
    float2 vs = *xs;   // coalesced: wave covers the full 256B row
    float2 vd = *xd;

    float* af = agg_f + (size_t)d * D + 2 * lane;
    float* ar = agg_r + (size_t)s * D + 2 * lane;
    // HW f32 atomics (no return -> STOREcnt path, resolves in L2)
    unsafeAtomicAdd(af + 0, vs.x);
    unsafeAtomicAdd(af + 1, vs.y);
    unsafeAtomicAdd(ar + 0, vd.x);
    unsafeAtomicAdd(ar + 1, vd.y);

    if (do_count) {
        if (lane == 0)      unsafeAtomicAdd(cnt_f + d, 1.0f);
        else if (lane == 1) unsafeAtomicAdd(cnt_r + s, 1.0f);
    }
}

// ---------------------------------------------------------------------------
// Fused layer GEMM via V_WMMA_F32_16X16X4_F32:
//   out = mean_f @ Wl.T + mean_r @ WlT.T + x @ (Wr+WrT).T + (b+bT)
// Single K=192 GEMM with stacked A = [mean_f | mean_r | x] and stacked B
// (in LDS, float2-interleaved over K so each step is one ds b64 read).
// One wave -> 16 rows x 64 cols of output (4 accumulator tiles).
// ---------------------------------------------------------------------------
__device__ __forceinline__ v8f wmma4(v2f a, v2f b, v8f c) {
    // 8 args: (neg_a, A, neg_b, B, c_mod, C, reuse_a, reuse_b)
    return __builtin_amdgcn_wmma_f32_16x16x4_f32(
        false, a, false, b, (short)0, c, false, false);
}

__global__ __launch_bounds__(256) void sage_gemm(
    const float* __restrict__ agg_f,
    const float* __restrict__ agg_r,
    const float* __restrict__ cnt_f,
    const float* __restrict__ cnt_r,
    const float* __restrict__ xin,
    const float* __restrict__ Wl,
    const float* __restrict__ WlT,
    const float* __restrict__ Wr,
    const float* __restrict__ WrT,
    const float* __restrict__ bias,
    const float* __restrict__ biasT,
    float* __restrict__ out)
{
    // B matrix: Bstk[k][n] = k<64 ? Wl[n][k] : k<128 ? WlT[n][k-64]
    //                               : Wr[n][k-128] + WrT[n][k-128]
    // Stored float2-interleaved: lds[(k>>1)*128 + n*2 + (k&1)]
    __shared__ float ldsB[192 * 64];  // 48 KB

    for (int idx = threadIdx.x; idx < 192 * 64; idx += 256) {
        int k = idx >> 6;   // 0..191
        int n = idx & 63;   // output column
        float v;
        if (k < 64)        v = Wl[n * D + k];
        else if (k < 128)  v = WlT[n * D + (k - 64)];
        else               v = Wr[n * D + (k - 128)] + WrT[n * D + (k - 128)];
        ldsB[((k >> 1) << 7) + (n << 1) + (k & 1)] = v;
    }
    __syncthreads();

    int lane = threadIdx.x & 31;
    int wave = threadIdx.x >> 5;
    int tile = blockIdx.x * 8 + wave;
    const int ntiles = (N_NODES + 15) / 16;
    if (tile >= ntiles) return;

    int rowbase = tile * 16;
    int nl = lane & 15;
    int arow = rowbase + nl;                         // A-row this lane feeds
    int arowc = arow < N_NODES ? arow : N_NODES - 1; // clamp (N%16==0 anyway)

    // counts are small exact integers: fast v_rcp is plenty for the mean
    float invf = __builtin_amdgcn_rcpf(fmaxf(cnt_f[arowc], 1.0f));
    float invr = __builtin_amdgcn_rcpf(fmaxf(cnt_r[arowc], 1.0f));

    const float* aseg[3] = { agg_f + (size_t)arowc * D,
                             agg_r + (size_t)arowc * D,
                             xin   + (size_t)arowc * D };
    const float  ascl[3] = { invf, invr, 1.0f };

    // A fragment K-split across lane halves (ISA 16x4 f32 layout)
    int khalf = (lane < 16) ? 0 : 2;

    v8f c0 = {}, c1 = {}, c2 = {}, c3 = {};

#pragma unroll
    for (int seg = 0; seg < 3; ++seg) {
        const float* ap = aseg[seg];
        float sc = ascl[seg];
#pragma unroll 4
        for (int kk = 0; kk < 16; ++kk) {
            int kloc = kk * 4 + khalf;           // k within this segment
            int kglob = seg * 64 + kloc;         // stacked k
            v2f a = { ap[kloc] * sc, ap[kloc + 1] * sc };
            const float* bb = ldsB + ((kglob >> 1) << 7) + (nl << 1);
            v2f b0 = *(const v2f*)(bb);
            v2f b1 = *(const v2f*)(bb + 32);
            v2f b2 = *(const v2f*)(bb + 64);
            v2f b3 = *(const v2f*)(bb + 96);
            c0 = wmma4(a, b0, c0);
            c1 = wmma4(a, b1, c1);
            c2 = wmma4(a, b2, c2);
            c3 = wmma4(a, b3, c3);
        }
    }

    // Epilogue: C/D layout -> VGPR v holds rows v (lanes 0-15) / v+8 (16-31)
    int rowoff = (lane < 16) ? 0 : 8;
#pragma unroll
    for (int ct = 0; ct < 4; ++ct) {
        int j = ct * 16 + nl;
        float bs = bias[j] + biasT[j];
        v8f c = (ct == 0) ? c0 : (ct == 1) ? c1 : (ct == 2) ? c2 : c3;
#pragma unroll
        for (int v = 0; v < 8; ++v) {
            int row = rowbase + rowoff + v;
            if (row < N_NODES)
                out[(size_t)row * D + j] = c[v] + bs;
        }
    }
}

// ---------------------------------------------------------------------------
extern "C" void kernel_launch(void* const* d_in, const int* in_sizes, int n_in,
                              void* d_out, int out_size, void* d_ws, size_t ws_size,
                              hipStream_t stream) {
    (void)in_sizes; (void)n_in; (void)out_size; (void)ws_size;

    const float* x    = (const float*)d_in[0];
    // edge_index: JAX default config has x64 disabled -> int32 on device
    const int*   ei   = (const int*)d_in[1];
    const float* Wl1  = (const float*)d_in[2];
    const float* Wr1  = (const float*)d_in[3];
    const float* b1   = (const float*)d_in[4];
    const float* WlT1 = (const float*)d_in[5];
    const float* WrT1 = (const float*)d_in[6];
    const float* bT1  = (const float*)d_in[7];
    const float* Wl2  = (const float*)d_in[8];
    const float* Wr2  = (const float*)d_in[9];
    const float* b2   = (const float*)d_in[10];
    const float* WlT2 = (const float*)d_in[11];
    const float* WrT2 = (const float*)d_in[12];
    const float* bT2  = (const float*)d_in[13];

    const int* src = ei;             // edge_index[0]
    const int* dst = ei + N_EDGES;   // edge_index[1]

    float* ws    = (float*)d_ws;
    float* agg_f = ws;
    float* agg_r = agg_f + (size_t)N_NODES * D;
    float* cnt_f = agg_r + (size_t)N_NODES * D;
    float* cnt_r = cnt_f + N_NODES;
    float* out   = (float*)d_out;    // also holds layer-1 activations h

    const size_t agg_bytes = 2 * (size_t)N_NODES * D * sizeof(float);
    const size_t all_bytes = agg_bytes + 2 * (size_t)N_NODES * sizeof(float);

    dim3 sblk(256);
    dim3 sgrid((unsigned)(((long long)N_EDGES + 7) / 8));   // 8 edges/block
    const int ntiles = (N_NODES + 15) / 16;
    dim3 gblk(256);
    dim3 ggrid((unsigned)((ntiles + 7) / 8));

    // ---- layer 1 ----
    hipMemsetAsync(d_ws, 0, all_bytes, stream);   // agg_f, agg_r, cnt_f, cnt_r
    sage_scatter<<<sgrid, sblk, 0, stream>>>(x, src, dst, agg_f, agg_r,
                                             cnt_f, cnt_r, /*do_count=*/1);
    sage_gemm<<<ggrid, gblk, 0, stream>>>(agg_f, agg_r, cnt_f, cnt_r, x,
                                          Wl1, WlT1, Wr1, WrT1, b1, bT1, out);

    // ---- layer 2 (counts reused; h lives in d_out) ----
    hipMemsetAsync(d_ws, 0, agg_bytes, stream);   // agg_f, agg_r only
    sage_scatter<<<sgrid, sblk, 0, stream>>>(out, src, dst, agg_f, agg_r,
                                             cnt_f, cnt_r, /*do_count=*/0);
    sage_gemm<<<ggrid, gblk, 0, stream>>>(agg_f, agg_r, cnt_f, cnt_r, out,
                                          Wl2, WlT2, Wr2, WrT2, b2, bT2, out);
}